// GNSMsg_26594437497097
// MI455X (gfx1250) — compile-verified
//
#include <hip/hip_runtime.h>
#include <math.h>

// ---------------------------------------------------------------------------
// Problem constants (match reference)
// ---------------------------------------------------------------------------
#define B_    16
#define N_    512
#define D_    10
#define EALL_ 130816          // N*(N-1)/2
#define ROWS_ (B_ * N_)       // 8192
#define SROW_ 34              // LDS activation row stride (even + conflict-free)

typedef __attribute__((ext_vector_type(2))) float v2f;
typedef __attribute__((ext_vector_type(8))) float v8f;

// weight column strides (K4 + 2: even for b64 alignment, conflict-free banks)
#define WS16 18
#define WS24 26
#define WS32 34

// ---------------------------------------------------------------------------
// WMMA tile: fp32 16x16 via V_WMMA_F32_16X16X4_F32.
// Activations sXin: row-major [16][SROW_]. Weights sWt: column-major
// [ncols][WS] so each B fragment is one aligned 8-byte LDS load.
//   A 16x4 frag: lanes 0-15 M=lane (V0=K0,V1=K1); lanes 16-31 (V0=K2,V1=K3)
//   B 4x16 frag: lanes 0-15 K=0,1; lanes 16-31 K=2,3 ; col = lane&15
//   C/D:         VGPR r -> row r + 8*(lane>>4), col = lane&15
// ---------------------------------------------------------------------------
template <int K4, int WS>
__device__ __forceinline__ v8f mlp_tile(const float* sXin, const float* sWt,
                                        const float* sB, int ntile, int lane)
{
    const int col  = lane & 15;
    const int hi   = lane >> 4;
    const int ocol = ntile * 16 + col;
    const float bv = sB[ocol];
    v8f c;
#pragma unroll
    for (int r = 0; r < 8; ++r) c[r] = bv;

#pragma unroll
    for (int kb = 0; kb < K4; kb += 4) {
        const int kk = kb + 2 * hi;                      // always even
        const v2f a  = *(const v2f*)&sXin[col * SROW_ + kk];
        const v2f bf = *(const v2f*)&sWt[ocol * WS + kk];
        c = __builtin_amdgcn_wmma_f32_16x16x4_f32(false, a, false, bf,
                                                  (short)0, c, false, false);
    }
    return c;
}

__device__ __forceinline__ void store_tile(float* sH, v8f c, int ntile, int lane)
{
    const int col = ntile * 16 + (lane & 15);
    const int hi  = lane >> 4;
#pragma unroll
    for (int r = 0; r < 8; ++r) sH[(r + 8 * hi) * SROW_ + col] = c[r];
}

// LayerNorm(32) * g + be, then LeakyReLU(0.1); sH -> sX (rows 0..15)
__device__ __forceinline__ void ln_lrelu(const float* sH, float* sX,
                                         const float* g, const float* be, int lane)
{
    if (lane < 16) {
        const int R = lane;
        float sum = 0.0f;
#pragma unroll
        for (int c = 0; c < 32; ++c) sum += sH[R * SROW_ + c];
        const float mu = sum * (1.0f / 32.0f);
        float var = 0.0f;
#pragma unroll
        for (int c = 0; c < 32; ++c) { float d = sH[R * SROW_ + c] - mu; var += d * d; }
        var *= (1.0f / 32.0f);
        const float inv = rsqrtf(var + 1e-5f);
#pragma unroll
        for (int c = 0; c < 32; ++c) {
            float y = (sH[R * SROW_ + c] - mu) * inv * g[c] + be[c];
            sX[R * SROW_ + c] = (y >= 0.0f) ? y : 0.1f * y;
        }
    }
}

// ---------------------------------------------------------------------------
// k_init: state from V0, zero m/deg/counter              grid 32x256
// ---------------------------------------------------------------------------
__global__ __launch_bounds__(256) void k_init(const float* __restrict__ V0,
                                              float* v, float* th, float* m,
                                              int* deg, int* nact)
{
    const int gid = blockIdx.x * 256 + threadIdx.x;   // < ROWS_
    if (gid == 0) *nact = 0;
    v[gid]  = V0[gid * 2 + 0];
    th[gid] = V0[gid * 2 + 1];
    deg[gid] = 0;
#pragma unroll
    for (int d = 0; d < D_; ++d) m[gid * D_ + d] = 0.0f;
}

// ---------------------------------------------------------------------------
// k_compact: build active edge list (Line density ~1%)    grid 8176x256
// rec.x = (b<<18)|(i<<9)|j ; rec.y = e  (0xFFFFFFFF = invalid)
// ---------------------------------------------------------------------------
__global__ __launch_bounds__(256) void k_compact(const int* __restrict__ Line,
                                                 int* nact, uint2* recs,
                                                 int* deg, int cap)
{
    const int gid = blockIdx.x * 256 + threadIdx.x;   // < B_*EALL_
    const int b = gid / EALL_;
    const int e = gid - b * EALL_;
    if (Line[gid] != 0) {
        // invert upper-triangular flat index e -> (i, j), i < j
        const double tn = 2.0 * N_ - 1.0;
        int i = (int)((tn - sqrt(tn * tn - 8.0 * (double)e)) * 0.5);
        if (i < 0) i = 0;
        if (i > N_ - 2) i = N_ - 2;
        while (i > 0 && (i * (2 * N_ - i - 1)) / 2 > e) --i;
        while (((i + 1) * (2 * N_ - i - 2)) / 2 <= e) ++i;
        const int j = i + 1 + (e - (i * (2 * N_ - i - 1)) / 2);

        atomicAdd(&deg[b * N_ + i], 1);
        atomicAdd(&deg[b * N_ + j], 1);
        const int slot = atomicAdd(nact, 1);
        if (slot < cap) {
            uint2 rc;
            rc.x = ((unsigned)b << 18) | ((unsigned)i << 9) | (unsigned)j;
            rc.y = (unsigned)e;
            recs[slot] = rc;
        }
    }
}

// ---------------------------------------------------------------------------
// k_prep: Vc = v*exp(i*th); zero M_neigh                  grid 320x256
// ---------------------------------------------------------------------------
__global__ __launch_bounds__(256) void k_prep(const float* __restrict__ v,
                                              const float* __restrict__ th,
                                              float* vcr, float* vci, float* mneigh)
{
    const int gid = blockIdx.x * 256 + threadIdx.x;   // < ROWS_*D_
    mneigh[gid] = 0.0f;
    if (gid < ROWS_) {
        const float vv = v[gid], tt = th[gid];
        vcr[gid] = vv * cosf(tt);
        vci[gid] = vv * sinf(tt);
    }
}

// ---------------------------------------------------------------------------
// k_mismatch: Ic = Y*Vc ; Sc = Vc*conj(Ic) ; DP/DQ        grid 1024x256
// ---------------------------------------------------------------------------
__global__ __launch_bounds__(256) void k_mismatch(const float* __restrict__ Yr,
                                                  const float* __restrict__ Yi,
                                                  const float* __restrict__ vcr,
                                                  const float* __restrict__ vci,
                                                  const int* __restrict__ bus_type,
                                                  const float* __restrict__ P_set,
                                                  const float* __restrict__ Q_set,
                                                  float* DP, float* DQ)
{
    __shared__ float sVr[N_], sVi[N_];
    const int b     = blockIdx.x >> 6;            // 64 blocks per batch
    const int nbase = (blockIdx.x & 63) * 8;
    for (int idx = threadIdx.x; idx < N_; idx += 256) {
        sVr[idx] = vcr[b * N_ + idx];
        sVi[idx] = vci[b * N_ + idx];
    }
    __syncthreads();

    const int wv   = threadIdx.x >> 5;
    const int lane = threadIdx.x & 31;
    const int n    = nbase + wv;
    const int row  = b * N_ + n;
    const float* yr = Yr + (size_t)row * N_;
    const float* yi = Yi + (size_t)row * N_;

    float icr = 0.0f, ici = 0.0f;
#pragma unroll 4
    for (int t = 0; t < 16; ++t) {
        const int j = lane + 32 * t;
        const float a = yr[j], q = yi[j], pr = sVr[j], pi = sVi[j];
        icr += a * pr - q * pi;
        ici += a * pi + q * pr;
    }
#pragma unroll
    for (int off = 16; off > 0; off >>= 1) {
        icr += __shfl_xor(icr, off, 32);
        ici += __shfl_xor(ici, off, 32);
    }
    if (lane == 0) {
        const float pr = sVr[n], pi = sVi[n];
        const float scr = pr * icr + pi * ici;
        const float sci = pi * icr - pr * ici;
        const int bt = bus_type[row];
        DP[row] = (bt == 1) ? 0.0f : (P_set[row] - scr);
        DQ[row] = (bt == 1 || bt == 2) ? 0.0f : (Q_set[row] - sci);
    }
}

// ---------------------------------------------------------------------------
// k_edge: WMMA edge MLP (13->32->32->10) over active edges; atomic scatter.
// Persistent: 512 blocks x 1 wave; weights staged transposed in LDS once.
// ---------------------------------------------------------------------------
__global__ __launch_bounds__(32) void k_edge(const uint2* __restrict__ recs,
                                             const int* __restrict__ nact_p, int cap,
                                             const float* __restrict__ m,
                                             const float* __restrict__ Ysr,
                                             const float* __restrict__ Ysi,
                                             const float* __restrict__ Yc,
                                             const float* W1, const float* b1,
                                             const float* g1, const float* be1,
                                             const float* W2, const float* b2,
                                             const float* g2, const float* be2,
                                             const float* W3, const float* b3,
                                             float* mneigh)
{
    __shared__ alignas(16) float sW1t[32 * WS16];   // [col][k], K 13->16 pad
    __shared__ alignas(16) float sW2t[32 * WS32];
    __shared__ alignas(16) float sW3t[16 * WS32];   // cols 10->16 pad
    __shared__ float sB1[32], sB2[32], sB3[16];
    __shared__ float sG1[32], sBE1[32], sG2[32], sBE2[32];
    __shared__ alignas(16) float sX[16 * SROW_];
    __shared__ alignas(16) float sH[16 * SROW_];
    __shared__ uint2 srec[16];
    const int lane = threadIdx.x;

    int count = *nact_p;
    if (count > cap) count = cap;
    const int ntiles = (count + 15) >> 4;
    if (ntiles == 0) return;

    // stage weights transposed + zero-padded
    for (int idx = lane; idx < 32 * 16; idx += 32) {
        const int c = idx >> 4, k = idx & 15;
        sW1t[c * WS16 + k] = (k < 13) ? W1[k * 32 + c] : 0.0f;
    }
    for (int idx = lane; idx < 32 * 32; idx += 32) {
        const int c = idx >> 5, k = idx & 31;
        sW2t[c * WS32 + k] = W2[k * 32 + c];
    }
    for (int idx = lane; idx < 16 * 32; idx += 32) {
        const int c = idx >> 5, k = idx & 31;
        sW3t[c * WS32 + k] = (c < D_) ? W3[k * D_ + c] : 0.0f;
    }
    sB1[lane] = b1[lane];  sB2[lane] = b2[lane];
    sG1[lane] = g1[lane];  sBE1[lane] = be1[lane];
    sG2[lane] = g2[lane];  sBE2[lane] = be2[lane];
    if (lane < 16) sB3[lane] = (lane < D_) ? b3[lane] : 0.0f;
    __syncthreads();

    for (int tile = blockIdx.x; tile < ntiles; tile += gridDim.x) {
        if (lane < 16) {
            const int idx = tile * 16 + lane;
            uint2 rc;
            if (idx < count) rc = recs[idx];
            else { rc.x = 0u; rc.y = 0xFFFFFFFFu; }
            srec[lane] = rc;
        }
        __syncthreads();

        // gather X = [m[b, j, :], Ysr, Ysi, Yc, 0, 0, 0]  (16 rows x 16 cols)
        for (int idx = lane; idx < 256; idx += 32) {
            const int r = idx >> 4, c = idx & 15;
            const uint2 rc = srec[r];
            float val = 0.0f;
            if (rc.y != 0xFFFFFFFFu) {
                const int bb = (int)(rc.x >> 18);
                const int jj = (int)(rc.x & 511u);
                const int e  = (int)rc.y;
                if (c < D_)       val = m[((bb << 9) + jj) * D_ + c];
                else if (c == 10) val = Ysr[bb * EALL_ + e];
                else if (c == 11) val = Ysi[bb * EALL_ + e];
                else if (c == 12) val = Yc [bb * EALL_ + e];
            }
            sX[r * SROW_ + c] = val;
        }
        __syncthreads();

        v8f c0 = mlp_tile<16, WS16>(sX, sW1t, sB1, 0, lane);
        v8f c1 = mlp_tile<16, WS16>(sX, sW1t, sB1, 1, lane);
        store_tile(sH, c0, 0, lane); store_tile(sH, c1, 1, lane);
        __syncthreads();
        ln_lrelu(sH, sX, sG1, sBE1, lane);
        __syncthreads();

        c0 = mlp_tile<32, WS32>(sX, sW2t, sB2, 0, lane);
        c1 = mlp_tile<32, WS32>(sX, sW2t, sB2, 1, lane);
        store_tile(sH, c0, 0, lane); store_tile(sH, c1, 1, lane);
        __syncthreads();
        ln_lrelu(sH, sX, sG2, sBE2, lane);
        __syncthreads();

        v8f cph = mlp_tile<32, WS32>(sX, sW3t, sB3, 0, lane);

        // scatter-add phi into M_neigh at both endpoints
        const int col = lane & 15, hi = lane >> 4;
        if (col < D_) {
#pragma unroll
            for (int r = 0; r < 8; ++r) {
                const int rr = r + 8 * hi;
                const uint2 rc = srec[rr];
                if (rc.y != 0xFFFFFFFFu) {
                    const int bb = (int)(rc.x >> 18);
                    const int ii = (int)((rc.x >> 9) & 511u);
                    const int jj = (int)(rc.x & 511u);
                    const float val = cph[r];
                    unsafeAtomicAdd(&mneigh[((bb << 9) + ii) * D_ + col], val);
                    unsafeAtomicAdd(&mneigh[((bb << 9) + jj) * D_ + col], val);
                }
            }
        }
        __syncthreads();
    }
}

// ---------------------------------------------------------------------------
// k_feats: assemble 24-dim node features                   grid 32x256
// ---------------------------------------------------------------------------
__global__ __launch_bounds__(256) void k_feats(const float* __restrict__ v,
                                               const float* __restrict__ th,
                                               const float* __restrict__ DP,
                                               const float* __restrict__ DQ,
                                               const float* __restrict__ m,
                                               const float* __restrict__ mneigh,
                                               const int* __restrict__ deg,
                                               float* feats)
{
    const int row = blockIdx.x * 256 + threadIdx.x;
    const float A = 1.0f / fmaxf((float)deg[row], 1.0f);
    float* f = feats + row * 24;
    f[0] = v[row]; f[1] = th[row]; f[2] = DP[row]; f[3] = DQ[row];
#pragma unroll
    for (int d = 0; d < D_; ++d) {
        f[4 + d]  = m[row * D_ + d];
        f[14 + d] = mneigh[row * D_ + d] * A;
    }
}

// ---------------------------------------------------------------------------
// k_node3: fused theta/v/m MLPs (24->32->32->{1,1,10}); persistent blocks,
// weights for the three heads staged transposed in LDS once per block.
// ---------------------------------------------------------------------------
__global__ __launch_bounds__(32) void k_node3(const float* __restrict__ feats,
    const float* tW1, const float* tb1, const float* tg1, const float* tbe1,
    const float* tW2, const float* tb2, const float* tg2, const float* tbe2,
    const float* tW3, const float* tb3,
    const float* vW1, const float* vb1, const float* vg1, const float* vbe1,
    const float* vW2, const float* vb2, const float* vg2, const float* vbe2,
    const float* vW3, const float* vb3,
    const float* mW1, const float* mb1, const float* mg1, const float* mbe1,
    const float* mW2, const float* mb2, const float* mg2, const float* mbe2,
    const float* mW3, const float* mb3,
    float* __restrict__ dth, float* __restrict__ dv, float* __restrict__ dmraw)
{
    __shared__ alignas(16) float sW1t[3][32 * WS24];
    __shared__ alignas(16) float sW2t[3][32 * WS32];
    __shared__ alignas(16) float sW3t[3][16 * WS32];
    __shared__ float sB1[3][32], sB2[3][32], sB3[3][16];
    __shared__ float sG1[3][32], sBE1[3][32], sG2[3][32], sBE2[3][32];
    __shared__ alignas(16) float sF[16 * SROW_];
    __shared__ alignas(16) float sX[16 * SROW_];
    __shared__ alignas(16) float sH[16 * SROW_];
    const int lane = threadIdx.x;

    const float* W1p[3]  = {tW1, vW1, mW1};
    const float* W2p[3]  = {tW2, vW2, mW2};
    const float* W3p[3]  = {tW3, vW3, mW3};
    const float* b1p[3]  = {tb1, vb1, mb1};
    const float* b2p[3]  = {tb2, vb2, mb2};
    const float* b3p[3]  = {tb3, vb3, mb3};
    const float* g1p[3]  = {tg1, vg1, mg1};
    const float* g2p[3]  = {tg2, vg2, mg2};
    const float* be1p[3] = {tbe1, vbe1, mbe1};
    const float* be2p[3] = {tbe2, vbe2, mbe2};

#pragma unroll
    for (int s = 0; s < 3; ++s) {
        const int dout = (s == 2) ? D_ : 1;
        for (int idx = lane; idx < 32 * 24; idx += 32) {
            const int c = idx / 24, k = idx - c * 24;
            sW1t[s][c * WS24 + k] = W1p[s][k * 32 + c];
        }
        for (int idx = lane; idx < 32 * 32; idx += 32) {
            const int c = idx >> 5, k = idx & 31;
            sW2t[s][c * WS32 + k] = W2p[s][k * 32 + c];
        }
        for (int idx = lane; idx < 16 * 32; idx += 32) {
            const int c = idx >> 5, k = idx & 31;
            sW3t[s][c * WS32 + k] = (c < dout) ? W3p[s][k * dout + c] : 0.0f;
        }
        sB1[s][lane] = b1p[s][lane];   sB2[s][lane] = b2p[s][lane];
        sG1[s][lane] = g1p[s][lane];   sBE1[s][lane] = be1p[s][lane];
        sG2[s][lane] = g2p[s][lane];   sBE2[s][lane] = be2p[s][lane];
        if (lane < 16) sB3[s][lane] = (lane < dout) ? b3p[s][lane] : 0.0f;
    }
    __syncthreads();

    for (int tile = blockIdx.x; tile < ROWS_ / 16; tile += gridDim.x) {
        for (int idx = lane; idx < 16 * 24; idx += 32) {
            const int r = idx / 24, c = idx - r * 24;
            sF[r * SROW_ + c] = feats[(tile * 16 + r) * 24 + c];
        }
        __syncthreads();

#pragma unroll
        for (int s = 0; s < 3; ++s) {
            v8f c0 = mlp_tile<24, WS24>(sF, sW1t[s], sB1[s], 0, lane);
            v8f c1 = mlp_tile<24, WS24>(sF, sW1t[s], sB1[s], 1, lane);
            store_tile(sH, c0, 0, lane); store_tile(sH, c1, 1, lane);
            __syncthreads();
            ln_lrelu(sH, sX, sG1[s], sBE1[s], lane);
            __syncthreads();

            c0 = mlp_tile<32, WS32>(sX, sW2t[s], sB2[s], 0, lane);
            c1 = mlp_tile<32, WS32>(sX, sW2t[s], sB2[s], 1, lane);
            store_tile(sH, c0, 0, lane); store_tile(sH, c1, 1, lane);
            __syncthreads();
            ln_lrelu(sH, sX, sG2[s], sBE2[s], lane);
            __syncthreads();

            v8f c2 = mlp_tile<32, WS32>(sX, sW3t[s], sB3[s], 0, lane);

            const int col = lane & 15, hi = lane >> 4;
            if (s == 2) {
                if (col < D_) {
#pragma unroll
                    for (int r = 0; r < 8; ++r)
                        dmraw[(tile * 16 + r + 8 * hi) * D_ + col] = c2[r];
                }
            } else {
                float* o = (s == 0) ? dth : dv;
                if (col == 0) {
#pragma unroll
                    for (int r = 0; r < 8; ++r)
                        o[tile * 16 + r + 8 * hi] = c2[r];
                }
            }
            __syncthreads();
        }
    }
}

// ---------------------------------------------------------------------------
// k_update: tanh+LN message update, clipped masked v/th update
// ---------------------------------------------------------------------------
__global__ __launch_bounds__(256) void k_update(const int* __restrict__ bus_type,
                                                const float* __restrict__ dthb,
                                                const float* __restrict__ dvb,
                                                const float* __restrict__ dmraw,
                                                float* v, float* th, float* m,
                                                float* out, int write_out)
{
    const int row = blockIdx.x * 256 + threadIdx.x;
    const int bt = bus_type[row];
    const bool slack = (bt == 1), pv = (bt == 2);

    float t[D_];
    float mu = 0.0f;
#pragma unroll
    for (int d = 0; d < D_; ++d) { t[d] = tanhf(dmraw[row * D_ + d]); mu += t[d]; }
    mu *= (1.0f / D_);
    float var = 0.0f;
#pragma unroll
    for (int d = 0; d < D_; ++d) { const float dd = t[d] - mu; var += dd * dd; }
    var *= (1.0f / D_);
    const float inv = rsqrtf(var + 1e-5f);
#pragma unroll
    for (int d = 0; d < D_; ++d) m[row * D_ + d] += (t[d] - mu) * inv;

    float dth = slack ? 0.0f : dthb[row];
    dth = fminf(0.3f, fmaxf(-0.3f, dth));
    const float vv = v[row];
    const float va = fabsf(vv);
    float dv = (slack || pv) ? 0.0f : dvb[row];
    dv = fminf(0.1f * va, fmaxf(-0.1f * va, dv));

    const float nv  = vv + dv;
    const float nth = th[row] + dth;
    v[row] = nv; th[row] = nth;
    if (write_out) { out[row * 2 + 0] = nv; out[row * 2 + 1] = nth; }
}

// ---------------------------------------------------------------------------
// host launcher
// ---------------------------------------------------------------------------
extern "C" void kernel_launch(void* const* d_in, const int* in_sizes, int n_in,
                              void* d_out, int out_size, void* d_ws, size_t ws_size,
                              hipStream_t stream)
{
    (void)in_sizes; (void)n_in; (void)out_size;
    const int*   bus_type = (const int*)d_in[0];
    const int*   Line     = (const int*)d_in[1];
    const float* Ysr      = (const float*)d_in[2];
    const float* Ysi      = (const float*)d_in[3];
    const float* Yc       = (const float*)d_in[4];
    const float* Yr       = (const float*)d_in[5];
    const float* Yi       = (const float*)d_in[6];
    const float* P_set    = (const float*)d_in[7];
    const float* Q_set    = (const float*)d_in[8];
    const float* V0       = (const float*)d_in[9];
    auto F = [&](int i) { return (const float*)d_in[i]; };

    // workspace layout
    char*  w      = (char*)d_ws;
    float* v_s    = (float*)w;
    float* th_s   = v_s + ROWS_;
    float* m_s    = th_s + ROWS_;                 // ROWS_*D_
    float* vcr    = m_s + ROWS_ * D_;
    float* vci    = vcr + ROWS_;
    float* dps    = vci + ROWS_;
    float* dqs    = dps + ROWS_;
    float* mneigh = dqs + ROWS_;                  // ROWS_*D_
    float* feats  = mneigh + ROWS_ * D_;          // ROWS_*24
    float* dth_s  = feats + ROWS_ * 24;
    float* dv_s   = dth_s + ROWS_;
    float* dmraw  = dv_s + ROWS_;                 // ROWS_*D_
    int*   deg    = (int*)(dmraw + ROWS_ * D_);
    int*   nact   = deg + ROWS_;
    size_t used   = (size_t)((char*)(nact + 1) - w);
    used = (used + 15) & ~(size_t)15;
    uint2* recs   = (uint2*)(w + used);
    long capL = ((long)ws_size - (long)used) / (long)sizeof(uint2);
    if (capL < 0) capL = 0;
    if (capL > (long)B_ * EALL_) capL = (long)B_ * EALL_;
    const int cap = (int)capL;

    k_init<<<ROWS_ / 256, 256, 0, stream>>>(V0, v_s, th_s, m_s, deg, nact);
    k_compact<<<(B_ * EALL_) / 256, 256, 0, stream>>>(Line, nact, recs, deg, cap);

    for (int k = 0; k < 4; ++k) {
        // params (jax pytree flatten: dict keys sorted; blocks edge,m,theta,v;
        // fields W1,W2,W3,b1,b2,b3,be1,be2,g1,g2), leading K dim sliced by k
        const float *eW1 = F(10) + k * 13 * 32, *eW2 = F(11) + k * 1024, *eW3 = F(12) + k * 320;
        const float *eb1 = F(13) + k * 32, *eb2 = F(14) + k * 32, *eb3 = F(15) + k * 10;
        const float *ebe1 = F(16) + k * 32, *ebe2 = F(17) + k * 32;
        const float *eg1 = F(18) + k * 32, *eg2 = F(19) + k * 32;

        const float *mW1 = F(20) + k * 24 * 32, *mW2 = F(21) + k * 1024, *mW3 = F(22) + k * 320;
        const float *mb1 = F(23) + k * 32, *mb2 = F(24) + k * 32, *mb3 = F(25) + k * 10;
        const float *mbe1 = F(26) + k * 32, *mbe2 = F(27) + k * 32;
        const float *mg1 = F(28) + k * 32, *mg2 = F(29) + k * 32;

        const float *tW1 = F(30) + k * 24 * 32, *tW2 = F(31) + k * 1024, *tW3 = F(32) + k * 32;
        const float *tb1 = F(33) + k * 32, *tb2 = F(34) + k * 32, *tb3 = F(35) + k * 1;
        const float *tbe1 = F(36) + k * 32, *tbe2 = F(37) + k * 32;
        const float *tg1 = F(38) + k * 32, *tg2 = F(39) + k * 32;

        const float *vW1 = F(40) + k * 24 * 32, *vW2 = F(41) + k * 1024, *vW3 = F(42) + k * 32;
        const float *vb1 = F(43) + k * 32, *vb2 = F(44) + k * 32, *vb3 = F(45) + k * 1;
        const float *vbe1 = F(46) + k * 32, *vbe2 = F(47) + k * 32;
        const float *vg1 = F(48) + k * 32, *vg2 = F(49) + k * 32;

        k_prep<<<(ROWS_ * D_) / 256, 256, 0, stream>>>(v_s, th_s, vcr, vci, mneigh);
        k_mismatch<<<ROWS_ / 8, 256, 0, stream>>>(Yr, Yi, vcr, vci, bus_type,
                                                  P_set, Q_set, dps, dqs);
        k_edge<<<512, 32, 0, stream>>>(recs, nact, cap, m_s, Ysr, Ysi, Yc,
                                       eW1, eb1, eg1, ebe1, eW2, eb2, eg2, ebe2,
                                       eW3, eb3, mneigh);
        k_feats<<<ROWS_ / 256, 256, 0, stream>>>(v_s, th_s, dps, dqs, m_s, mneigh,
                                                 deg, feats);
        k_node3<<<128, 32, 0, stream>>>(feats,
                                        tW1, tb1, tg1, tbe1, tW2, tb2, tg2, tbe2, tW3, tb3,
                                        vW1, vb1, vg1, vbe1, vW2, vb2, vg2, vbe2, vW3, vb3,
                                        mW1, mb1, mg1, mbe1, mW2, mb2, mg2, mbe2, mW3, mb3,
                                        dth_s, dv_s, dmraw);
        k_update<<<ROWS_ / 256, 256, 0, stream>>>(bus_type, dth_s, dv_s, dmraw,
                                                  v_s, th_s, m_s,
                                                  (float*)d_out, (k == 3) ? 1 : 0);
    }
}